// AugmentedConv_11957188952713
// MI455X (gfx1250) — compile-verified
//
#include <hip/hip_runtime.h>
#include <hip/hip_bf16.h>
#include <math.h>

typedef __attribute__((ext_vector_type(2))) float v2f;
typedef __attribute__((ext_vector_type(4))) float v4f;
typedef __attribute__((ext_vector_type(8))) float v8f;
typedef __attribute__((ext_vector_type(4))) int   v4i;

#define Wlen 2048
#define CIN  256
#define KK   768          /* Cin * 3 */
#define OUTC 512

/* ---- CDNA5 async global->LDS staging (ASYNCcnt path), with fallback ---- */
#if defined(__has_builtin)
#if __has_builtin(__builtin_amdgcn_global_load_async_to_lds_b128) && \
    __has_builtin(__builtin_amdgcn_s_wait_asynccnt)
#define HAVE_ASYNC_LDS 1
#endif
#endif

#if defined(HAVE_ASYNC_LDS)
#define ASYNC_COPY16(gsrc, ldst)                                   \
  __builtin_amdgcn_global_load_async_to_lds_b128(                  \
      (__attribute__((address_space(1))) v4i*)(gsrc),              \
      (__attribute__((address_space(3))) v4i*)(ldst), 0, 0)
#define ASYNC_WAIT() __builtin_amdgcn_s_wait_asynccnt(0)
#else
#define ASYNC_COPY16(gsrc, ldst) (*(v4f*)(ldst) = *(const v4f*)(gsrc))
#define ASYNC_WAIT() ((void)0)
#endif

/* wave32 XOR swizzle within 16-lane halves (masks < 16 never cross lane 16) */
#define SWZ_XOR(v, mask) \
  __int_as_float(__builtin_amdgcn_ds_swizzle(__float_as_int(v), (((mask) << 10) | 0x1f)))

__device__ __forceinline__ v8f wmma4(v2f a, v2f b, v8f c) {
  /* D = A(16x4 f32) * B(4x16 f32) + C(16x16 f32), full fp32 precision */
  return __builtin_amdgcn_wmma_f32_16x16x4_f32(false, a, false, b, (short)0, c, false, false);
}

__device__ __forceinline__ v8f vzero8() {
  v8f z = {0.f, 0.f, 0.f, 0.f, 0.f, 0.f, 0.f, 0.f};
  return z;
}

/* ------------------------------------------------------------------ *
 * Kernel 1: fused conv1d (K=3, pad=1) for [w_conv(256); w_qkv(768)]  *
 * implicit GEMM, M=1024, N=2048, K=768.  Block = 4 waves sharing one *
 * 64-wide w-block; x patch staged in LDS (async B128) with halo so   *
 * the inner loop is branchless ds_load + V_WMMA_F32_16X16X4_F32.     *
 * ------------------------------------------------------------------ */
__global__ void __launch_bounds__(128) conv_qkv_kernel(
    const float* __restrict__ x, const float* __restrict__ w_conv,
    const float* __restrict__ b_conv, const float* __restrict__ w_qkv,
    const float* __restrict__ b_qkv, float* __restrict__ out,
    float* __restrict__ qkv) {
  /* 128 ci rows x (4 halo/pad + 64 interior + halo) ; stride 72 keeps 16B align */
  __shared__ float xs[128][72];

  const int tid  = threadIdx.x;
  const int lane = tid & 31;
  const int wid  = blockIdx.x * 4 + (tid >> 5);
  const int tilesPerB = 64 * 32;                 /* co-tiles x w-tiles */
  const int b   = wid / tilesPerB;
  const int rem = wid % tilesPerB;
  const int co0 = (rem & 63) * 16;               /* co-tile fastest -> block shares w0 */
  const int w0  = (rem >> 6) * 64;

  const int half = lane >> 4;
  const int m    = lane & 15;

  const bool isConv = (co0 < 256);
  const float* wbase = isConv ? (w_conv + (size_t)co0 * KK)
                              : (w_qkv + (size_t)(co0 - 256) * KK);
  const float* xb = x + (size_t)b * CIN * Wlen;

  v8f acc[4];
#pragma unroll
  for (int j = 0; j < 4; ++j) acc[j] = vzero8();

  for (int cc = 0; cc < 2; ++cc) {               /* ci chunks of 128 rows */
    const int cbase = cc * 128;
    __syncthreads();                             /* protect previous chunk reads */

    /* interior: 128 rows x 64 cols, B128 async copies (16B aligned both sides) */
#pragma unroll
    for (int i = 0; i < 16; ++i) {
      const int idx = tid + i * 128;
      const int r = idx >> 4, g = idx & 15;
      ASYNC_COPY16(xb + (size_t)(cbase + r) * Wlen + w0 + g * 4, &xs[r][4 + g * 4]);
    }
    /* halo columns (w0-1 and w0+64) with zero padding at image edges */
#pragma unroll
    for (int i = 0; i < 2; ++i) {
      const int idx  = tid + i * 128;
      const int r    = idx >> 1;
      const int side = idx & 1;
      const int gw   = side ? (w0 + 64) : (w0 - 1);
      xs[r][side ? 68 : 3] =
          ((unsigned)gw < (unsigned)Wlen) ? xb[(size_t)(cbase + r) * Wlen + gw] : 0.f;
    }
    ASYNC_WAIT();
    __syncthreads();

    /* K slice for this chunk: p in [cbase*3, cbase*3+384), step 4 */
    const int pbeg = cbase * 3;
    for (int p = pbeg; p < pbeg + 384; p += 4) {
      v2f a = *(const v2f*)(wbase + (size_t)m * KK + p + 2 * half);

      const int pd = p / 3;                      /* uniform -> SALU */
      const int pm = p % 3;
      int k0  = pm + 2 * half;                   /* per-lane fixup, no division */
      int ci0 = (pd - cbase) + (k0 >= 3 ? 1 : 0);
      k0 -= (k0 >= 3 ? 3 : 0);
      int k1  = k0 + 1;
      int ci1 = ci0 + (k1 == 3 ? 1 : 0);
      k1 = (k1 == 3 ? 0 : k1);

      const float* r0p = &xs[ci0][k0 + 3 + m];
      const float* r1p = &xs[ci1][k1 + 3 + m];
#pragma unroll
      for (int j = 0; j < 4; ++j) {
        v2f bb;
        bb.x = r0p[j * 16];
        bb.y = r1p[j * 16];
        acc[j] = wmma4(a, bb, acc[j]);
      }
    }
  }

#pragma unroll
  for (int r = 0; r < 8; ++r) {
    const int row = co0 + r + 8 * half;
    const float bias = isConv ? b_conv[row] : b_qkv[row - 256];
#pragma unroll
    for (int j = 0; j < 4; ++j) {
      const int col = w0 + j * 16 + m;
      const float val = acc[j][r] + bias;
      if (isConv)
        out[(size_t)b * OUTC * Wlen + (size_t)row * Wlen + col] = val;
      else
        qkv[(size_t)b * 768 * Wlen + (size_t)(row - 256) * Wlen + col] = val;
    }
  }
}

/* ------------------------------------------------------------------ *
 * Kernel 2: flash attention, one wave per (b, h, 16-query tile).     *
 * S = (Q*scale)^T K (K-dim = dkh = 32), streaming softmax over 128   *
 * key tiles, O += P V^T.  P goes D-layout -> LDS -> A-layout.        *
 * Writes the reference's faithful reshape directly.                  *
 * ------------------------------------------------------------------ */
__global__ void __launch_bounds__(128) attn_kernel(const float* __restrict__ qkv,
                                                   float* __restrict__ attn) {
  __shared__ float plds[4][16][17];
  const int lane  = threadIdx.x & 31;
  const int wslot = threadIdx.x >> 5;
  const int wid   = blockIdx.x * 4 + wslot;
  const int qt = wid & 127;
  const int h  = (wid >> 7) & 7;
  const int b  = wid >> 10;
  const int q0 = qt * 16;
  const int half = lane >> 4;
  const int m    = lane & 15;

  const float scale = 0.17677669529663689f; /* 1/sqrt(32) */
  const float* qb = qkv + ((size_t)b * 768 + (size_t)h * 32) * Wlen;
  const float* kb = qkv + ((size_t)b * 768 + 256 + (size_t)h * 32) * Wlen;
  const float* vb = qkv + ((size_t)b * 768 + 512 + (size_t)h * 32) * Wlen;

  /* preload Q as A fragments: A[m][kd] = q[kd][q0+m] * scale */
  v2f aq[8];
#pragma unroll
  for (int c = 0; c < 8; ++c) {
    const int kd = c * 4 + 2 * half;
    aq[c].x = qb[(size_t)kd * Wlen + q0 + m] * scale;
    aq[c].y = qb[(size_t)(kd + 1) * Wlen + q0 + m] * scale;
  }

  v8f o0 = vzero8(), o1 = vzero8();
  float mrun[8], lrun[8];
#pragma unroll
  for (int r = 0; r < 8; ++r) { mrun[r] = -1e30f; lrun[r] = 0.f; }

  for (int kblk = 0; kblk < 128; ++kblk) {
    const int k0 = kblk * 16;

    /* S tile: 8 WMMAs over the 32-deep head dim */
    v8f s = vzero8();
#pragma unroll
    for (int c = 0; c < 8; ++c) {
      const int kd = c * 4 + 2 * half;
      v2f bk;
      bk.x = kb[(size_t)kd * Wlen + k0 + m];
      bk.y = kb[(size_t)(kd + 1) * Wlen + k0 + m];
      s = wmma4(aq[c], bk, s);
    }

    /* streaming softmax: row stats live per (r, lane-half) */
    float alpha[8];
#pragma unroll
    for (int r = 0; r < 8; ++r) {
      float t = s[r];
      t = fmaxf(t, SWZ_XOR(t, 1));
      t = fmaxf(t, SWZ_XOR(t, 2));
      t = fmaxf(t, SWZ_XOR(t, 4));
      t = fmaxf(t, SWZ_XOR(t, 8));
      const float nm = fmaxf(mrun[r], t);
      alpha[r] = __expf(mrun[r] - nm);
      mrun[r]  = nm;
      const float p = __expf(s[r] - nm);
      s[r] = p;
      float rs = p;
      rs += SWZ_XOR(rs, 1);
      rs += SWZ_XOR(rs, 2);
      rs += SWZ_XOR(rs, 4);
      rs += SWZ_XOR(rs, 8);
      lrun[r] = lrun[r] * alpha[r] + rs;
    }
#pragma unroll
    for (int r = 0; r < 8; ++r) {
      o0[r] *= alpha[r];
      o1[r] *= alpha[r];
    }

    /* P: D-layout -> LDS row-major -> A-layout fragments */
#pragma unroll
    for (int r = 0; r < 8; ++r) plds[wslot][r + 8 * half][m] = s[r];
    __syncthreads();
    v2f ap[4];
#pragma unroll
    for (int c = 0; c < 4; ++c) {
      ap[c].x = plds[wslot][m][c * 4 + 2 * half];
      ap[c].y = plds[wslot][m][c * 4 + 2 * half + 1];
    }
    __syncthreads();

    /* O += P @ V^T  (B[krow][d] = v[d][k0+krow], contiguous pair loads) */
#pragma unroll
    for (int c = 0; c < 4; ++c) {
      const int krow = c * 4 + 2 * half;
      v2f bv0 = *(const v2f*)(vb + (size_t)m * Wlen + k0 + krow);
      v2f bv1 = *(const v2f*)(vb + (size_t)(16 + m) * Wlen + k0 + krow);
      o0 = wmma4(ap[c], bv0, o0);
      o1 = wmma4(ap[c], bv1, o1);
    }
  }

  /* normalize + faithful reshape: c = h*32 + (q>>6), w = (q&63)*32 + d */
#pragma unroll
  for (int r = 0; r < 8; ++r) {
    const int q    = q0 + r + 8 * half;
    const float nv = 1.0f / lrun[r];
    const int crow = h * 32 + (q >> 6);
    const int wcol = (q & 63) * 32;
    float* dst = attn + (size_t)b * 256 * Wlen + (size_t)crow * Wlen + wcol;
    dst[m]      = o0[r] * nv;
    dst[16 + m] = o1[r] * nv;
  }
}

/* ------------------------------------------------------------------ *
 * Kernel 3: 1x1 projection GEMM (256 x 2048 x 256 per batch) + bias. *
 * Block = 4 waves (o-tile fastest) sharing the attn B-tile staged in *
 * LDS via async B128; writes output channels 256..511.               *
 * ------------------------------------------------------------------ */
__global__ void __launch_bounds__(128) proj_kernel(const float* __restrict__ attn,
                                                   const float* __restrict__ w_attn,
                                                   const float* __restrict__ b_attn,
                                                   float* __restrict__ out) {
  __shared__ float bs[128][68];                  /* 128 K rows x 64 cols, 16B align */

  const int tid  = threadIdx.x;
  const int lane = tid & 31;
  const int wid  = blockIdx.x * 4 + (tid >> 5);
  const int tilesPerB = 16 * 32;                 /* o-tiles x w-tiles */
  const int b   = wid / tilesPerB;
  const int rem = wid % tilesPerB;
  const int o0  = (rem & 15) * 16;               /* o-tile fastest -> block shares w0 */
  const int w0  = (rem >> 4) * 64;

  const int half = lane >> 4;
  const int m    = lane & 15;

  const float* ab = attn + (size_t)b * 256 * Wlen;

  v8f acc[4];
#pragma unroll
  for (int j = 0; j < 4; ++j) acc[j] = vzero8();

  for (int cc = 0; cc < 2; ++cc) {               /* K chunks of 128 */
    const int cbase = cc * 128;
    __syncthreads();
#pragma unroll
    for (int i = 0; i < 16; ++i) {
      const int idx = tid + i * 128;
      const int r = idx >> 4, g = idx & 15;
      ASYNC_COPY16(ab + (size_t)(cbase + r) * Wlen + w0 + g * 4, &bs[r][g * 4]);
    }
    ASYNC_WAIT();
    __syncthreads();

    for (int p = 0; p < 128; p += 4) {
      v2f a = *(const v2f*)(w_attn + (size_t)(o0 + m) * 256 + cbase + p + 2 * half);
      const int pr = p + 2 * half;
#pragma unroll
      for (int j = 0; j < 4; ++j) {
        v2f bb;
        bb.x = bs[pr][j * 16 + m];
        bb.y = bs[pr + 1][j * 16 + m];
        acc[j] = wmma4(a, bb, acc[j]);
      }
    }
  }

#pragma unroll
  for (int r = 0; r < 8; ++r) {
    const int row = o0 + r + 8 * half;
    const float bias = b_attn[row];
#pragma unroll
    for (int j = 0; j < 4; ++j) {
      const int col = w0 + j * 16 + m;
      out[(size_t)b * OUTC * Wlen + (size_t)(256 + row) * Wlen + col] =
          acc[j][r] + bias;
    }
  }
}

extern "C" void kernel_launch(void* const* d_in, const int* in_sizes, int n_in,
                              void* d_out, int out_size, void* d_ws, size_t ws_size,
                              hipStream_t stream) {
  const float* x      = (const float*)d_in[0];
  const float* w_conv = (const float*)d_in[1];
  const float* b_conv = (const float*)d_in[2];
  const float* w_qkv  = (const float*)d_in[3];
  const float* b_qkv  = (const float*)d_in[4];
  const float* w_attn = (const float*)d_in[5];
  const float* b_attn = (const float*)d_in[6];
  float* out  = (float*)d_out;
  float* qkv  = (float*)d_ws;                    /* 8*768*2048 f32 = 50.3 MB */
  float* attn = qkv + (size_t)8 * 768 * Wlen;    /* 8*256*2048 f32 = 16.8 MB */

  dim3 blk(128);
  /* 16384 waves / 4 per block */
  conv_qkv_kernel<<<dim3(4096), blk, 0, stream>>>(x, w_conv, b_conv, w_qkv,
                                                  b_qkv, out, qkv);
  /* 8192 waves (b,h,qtile) / 4 per block */
  attn_kernel<<<dim3(2048), blk, 0, stream>>>(qkv, attn);
  /* 4096 waves / 4 per block */
  proj_kernel<<<dim3(1024), blk, 0, stream>>>(attn, w_attn, b_attn, out);
}